// FFTConv2d_85177791414304
// MI455X (gfx1250) — compile-verified
//
#include <hip/hip_runtime.h>

typedef __attribute__((ext_vector_type(2))) float v2f;
typedef __attribute__((ext_vector_type(4))) float v4f;
typedef __attribute__((ext_vector_type(8))) float v8f;

#define BATCH  8
#define CIN    32
#define COUT   32
#define HW     128
#define PLANE  (HW * HW)     // 16384 floats per channel plane
#define REGION 66            // crop of FFT full-conv that overlaps its support
#define XT     5             // ceil(66/16) x-tiles
#define SCALE  (1.0f/256.0f) // net ortho-norm fft*fft*ifft factor

// ---------------------------------------------------------------------------
// Phase 1: out[b,o,y,x] = bias[o] for the entire 8x32x128x128 tensor.
// ---------------------------------------------------------------------------
__global__ void __launch_bounds__(256) bias_fill_kernel(
    const float* __restrict__ bias, float* __restrict__ out) {
  int i = blockIdx.x * 256 + threadIdx.x;      // float4 index, 1048576 total
  int ch = (i >> 12) & (COUT - 1);             // (i / 4096) % 32
  float b = bias[ch];
  v4f v = {b, b, b, b};
  reinterpret_cast<v4f*>(out)[i] = v;
}

// ---------------------------------------------------------------------------
// Phase 2: overwrite the 66x66 live corner with bias + (1/256)*full-conv.
// Implicit GEMM, K enumerated as k = tap*32 + c  (tap = ky*3+kx):
//   - each K=4 WMMA step sits inside one tap -> (ky,kx) are uniform scalars
//   - activation address = per-tap base + c*PLANE (24-bit immediate offsets)
//   - out-of-tensor taps masked by a per-tap 0/1 float folded into B
// 72 chained V_WMMA_F32_16X16X4_F32, f32 in / f32 acc (reference precision).
// One wave32 per block; grid = (xtile + 5*otile, y, b).
// ---------------------------------------------------------------------------
__global__ void __launch_bounds__(32) fftconv_corner_wmma(
    const float* __restrict__ x,      // (8,32,128,128)
    const float* __restrict__ w,      // (32,32,3,3) flat: o*288 + c*9 + tap
    const float* __restrict__ bias,   // (32,)
    float* __restrict__ out) {        // (8,32,128,128)
  const int lane = threadIdx.x;       // 0..31
  const int half = lane >> 4;         // 0 or 1
  const int mn   = lane & 15;
  const int xt   = blockIdx.x % XT;
  const int ot   = blockIdx.x / XT;   // 0 or 1
  const int y    = blockIdx.y;        // 0..65
  const int b    = blockIdx.z;        // 0..7

  const int oA = ot * 16 + mn;        // A-matrix row (output channel) this lane feeds
  const int xg = xt * 16 + mn;        // B-matrix col (output x) this lane feeds

  const float* __restrict__ wrow = w + (size_t)oA * (CIN * 9);
  const float* __restrict__ xb   = x + (size_t)b * CIN * PLANE;

  v8f acc = {0.f, 0.f, 0.f, 0.f, 0.f, 0.f, 0.f, 0.f};

  #pragma unroll
  for (int tap = 0; tap < 9; ++tap) {
    const int ky = tap / 3;           // compile-time after unroll
    const int kx = tap % 3;
    const int row = 64 + y  - ky;     // uniform scalar; in [62,129]
    const int col = 64 + xg - kx;     // per-lane;       in [62,143]
    // Branch-free OOB handling: clamp address, zero via multiply.
    const float maskf = (row < HW && col < HW) ? 1.0f : 0.0f;
    const int rowc = row < HW ? row : HW - 1;
    const int colc = col < HW ? col : HW - 1;
    const float* __restrict__ xbase = xb + (rowc * HW + colc);
    const float* __restrict__ wbase = wrow + tap;

    #pragma unroll
    for (int c4 = 0; c4 < CIN / 4; ++c4) {
      const int c0 = c4 * 4 + 2 * half;   // this lane's two K slots (channels)
      v2f a, bm;
      a.x  = wbase[c0 * 9];               // w[oA][c0][tap]
      a.y  = wbase[c0 * 9 + 9];           // w[oA][c0+1][tap]
      bm.x = xbase[c0 * PLANE] * maskf;   // x[b][c0  ][row][col]
      bm.y = xbase[c0 * PLANE + PLANE] * maskf;
      acc = __builtin_amdgcn_wmma_f32_16x16x4_f32(false, a, false, bm,
                                                  (short)0, acc, false, false);
    }
  }

  // Store: lane holds D[M = r + 8*half][N = mn]; N is the output x position.
  const int xo = xt * 16 + mn;
  if (xo < REGION) {
    #pragma unroll
    for (int r = 0; r < 8; ++r) {
      const int o = ot * 16 + r + 8 * half;
      out[(((size_t)b * COUT + o) * HW + y) * HW + xo] = acc[r] * SCALE + bias[o];
    }
  }
}

// ---------------------------------------------------------------------------
// Launch: inputs are [x, weight, bias, stride(=1), padding(=1)].
// ---------------------------------------------------------------------------
extern "C" void kernel_launch(void* const* d_in, const int* in_sizes, int n_in,
                              void* d_out, int out_size, void* d_ws, size_t ws_size,
                              hipStream_t stream) {
  const float* x    = (const float*)d_in[0];
  const float* w    = (const float*)d_in[1];
  const float* bias = (const float*)d_in[2];
  float* out = (float*)d_out;

  // Phase 1: broadcast bias over entire output (16 MB, bandwidth-bound).
  const int n4 = BATCH * COUT * HW * HW / 4;      // 1048576 float4
  bias_fill_kernel<<<n4 / 256, 256, 0, stream>>>(bias, out);

  // Phase 2: WMMA conv over the 66x66 live corner.
  dim3 grid(XT * 2, REGION, BATCH);
  fftconv_corner_wmma<<<grid, 32, 0, stream>>>(x, w, bias, out);
}